// Matching_33165737459832
// MI455X (gfx1250) — compile-verified
//
#include <hip/hip_runtime.h>
#include <hip/hip_bf16.h>

typedef __attribute__((ext_vector_type(2))) float v2f;
typedef __attribute__((ext_vector_type(8))) float v8f;

#define BATCH 4
#define CH    128
#define HH    128
#define WW    128
#define HWSZ  (HH*WW)

// ---------------------------------------------------------------------------
// 1) Grouped ConvTranspose2d(2,2,k=4,s=2,p=1,groups=2)
// ---------------------------------------------------------------------------
__global__ void k_upsample(const float* __restrict__ flow, const float* __restrict__ upw,
                           float* __restrict__ flow_up) {
  int idx = blockIdx.x * blockDim.x + threadIdx.x;   // 131072
  if (idx >= BATCH*2*HWSZ) return;
  int x = idx & 127, y = (idx >> 7) & 127, g = (idx >> 14) & 1, b = idx >> 15;
  float acc = 0.f;
#pragma unroll
  for (int ky = 0; ky < 4; ++ky) {
    int ty = y + 1 - ky;
    if (ty & 1) continue;
    int iy = ty >> 1;
    if (iy < 0 || iy >= 64) continue;
#pragma unroll
    for (int kx = 0; kx < 4; ++kx) {
      int tx = x + 1 - kx;
      if (tx & 1) continue;
      int ix = tx >> 1;
      if (ix < 0 || ix >= 64) continue;
      acc += flow[((b*2 + g)*64 + iy)*64 + ix] * upw[g*16 + ky*4 + kx];
    }
  }
  flow_up[idx] = acc;
}

// ---------------------------------------------------------------------------
// 2) Bilinear warp with reflection (align_corners); H==W so coords simplify.
// ---------------------------------------------------------------------------
__global__ void k_warp(const float* __restrict__ f2, const float* __restrict__ flow_up,
                       float* __restrict__ out) {
  int b = blockIdx.y;
  int y = blockIdx.x;
  int x = threadIdx.x;                // 128 threads = one row
  float u0 = flow_up[((b*2 + 0)*HH + y)*WW + x];
  float u1 = flow_up[((b*2 + 1)*HH + y)*WW + x];
  float xp = (float)x + 0.05f * u1;
  float yp = (float)y + 0.05f * u0;
  xp = 127.0f - fabsf(fmodf(fabsf(xp), 254.0f) - 127.0f);
  yp = 127.0f - fabsf(fmodf(fabsf(yp), 254.0f) - 127.0f);
  float x0f = floorf(xp), y0f = floorf(yp);
  float wx = xp - x0f,   wy = yp - y0f;
  int x0 = (int)x0f; x0 = x0 < 0 ? 0 : (x0 > 127 ? 127 : x0);
  int x1 = x0 + 1 > 127 ? 127 : x0 + 1;
  int y0 = (int)y0f; y0 = y0 < 0 ? 0 : (y0 > 127 ? 127 : y0);
  int y1 = y0 + 1 > 127 ? 127 : y0 + 1;
  float w00 = (1.f-wx)*(1.f-wy), w01 = wx*(1.f-wy), w10 = (1.f-wx)*wy, w11 = wx*wy;
  const float* fb = f2 + (size_t)b * CH * HWSZ;
  float* ob = out + (size_t)b * CH * HWSZ + y*WW + x;
  int i00 = y0*WW + x0, i01 = y0*WW + x1, i10 = y1*WW + x0, i11 = y1*WW + x1;
  for (int c = 0; c < CH; ++c) {
    const float* fc = fb + c * HWSZ;
    ob[(size_t)c * HWSZ] = w00*fc[i00] + w01*fc[i01] + w10*fc[i10] + w11*fc[i11];
  }
}

// ---------------------------------------------------------------------------
// 3) 7x7 cost volume (mean over 128 ch) + leaky(0.01). Block = one (b,y) row.
// ---------------------------------------------------------------------------
__global__ void k_corr(const float* __restrict__ f1, const float* __restrict__ f2w,
                       float* __restrict__ corr) {
  __shared__ float s1[WW];
  __shared__ float s2[7 * WW];
  int b = blockIdx.y;
  int y = blockIdx.x;
  int x = threadIdx.x;                // 128 threads
  float acc[49];
#pragma unroll
  for (int d = 0; d < 49; ++d) acc[d] = 0.f;

  for (int c = 0; c < CH; ++c) {
    s1[x] = f1[(((size_t)b*CH + c)*HH + y)*WW + x];
#pragma unroll
    for (int dy = 0; dy < 7; ++dy) {
      int yy = y + dy - 3;
      s2[dy*WW + x] = (yy >= 0 && yy < HH)
          ? f2w[(((size_t)b*CH + c)*HH + yy)*WW + x] : 0.f;
    }
    __syncthreads();
    float v1 = s1[x];
#pragma unroll
    for (int dy = 0; dy < 7; ++dy) {
#pragma unroll
      for (int dx = 0; dx < 7; ++dx) {
        int xx = x + dx - 3;
        float v2 = (xx >= 0 && xx < WW) ? s2[dy*WW + xx] : 0.f;
        acc[dy*7 + dx] += v1 * v2;
      }
    }
    __syncthreads();
  }
#pragma unroll
  for (int d = 0; d < 49; ++d) {
    float v = acc[d] * (1.0f / 128.0f);
    v = v >= 0.f ? v : 0.01f * v;
    corr[(((size_t)b*49 + d)*HH + y)*WW + x] = v;
  }
}

// ---------------------------------------------------------------------------
// Weight reorder: w[Cout][Cin][3][3] -> wr[Cout][9][CinPad] (channel-inner,
// zero-padded). Makes A fragments single b64 loads and kills the Cin=49 tail.
// ---------------------------------------------------------------------------
template<int Cin, int Cout, int CinPad>
__global__ void k_wreorder(const float* __restrict__ w, float* __restrict__ wr) {
  int idx = blockIdx.x * blockDim.x + threadIdx.x;
  if (idx >= Cout * 9 * CinPad) return;
  int ci = idx % CinPad;
  int t  = (idx / CinPad) % 9;
  int co = idx / (9 * CinPad);
  wr[idx] = (ci < Cin) ? w[(co*Cin + ci)*9 + t] : 0.f;
}

// ---------------------------------------------------------------------------
// 4) 3x3 conv (pad 1): LDS-staged implicit GEMM, V_WMMA_F32_16X16X4_F32.
//    LDS layout channel-innermost: smem[pix][ci], pix = r*66 + p (3 rows of
//    64+2 halo, zero-padded).  B fragment (2 consecutive channels, same pixel)
//    = one ds_load_b64; A fragment (reordered weights) = one global b64.
//    CSTC chosen so 16 lanes (stride CSTC dwords) hit distinct LDS banks.
// ---------------------------------------------------------------------------
template<int CinPad, int CSTC>
__device__ __forceinline__ void step9(const float* __restrict__ wb,
                                      const float* __restrict__ sc,
                                      v8f& a0, v8f& a1, v8f& a2, v8f& a3) {
#pragma unroll
  for (int t = 0; t < 9; ++t) {
    const int pt = (t/3)*66 + (t%3);     // tap pixel offset (lane base excl.)
    v2f a  = *(const v2f*)(wb + t*CinPad);
    v2f b0 = *(const v2f*)(sc + (pt +  0)*CSTC);
    v2f b1 = *(const v2f*)(sc + (pt + 16)*CSTC);
    v2f b2 = *(const v2f*)(sc + (pt + 32)*CSTC);
    v2f b3 = *(const v2f*)(sc + (pt + 48)*CSTC);
    a0 = __builtin_amdgcn_wmma_f32_16x16x4_f32(false, a, false, b0, (short)0, a0, false, false);
    a1 = __builtin_amdgcn_wmma_f32_16x16x4_f32(false, a, false, b1, (short)0, a1, false, false);
    a2 = __builtin_amdgcn_wmma_f32_16x16x4_f32(false, a, false, b2, (short)0, a2, false, false);
    a3 = __builtin_amdgcn_wmma_f32_16x16x4_f32(false, a, false, b3, (short)0, a3, false, false);
  }
}

template<int Cin, int Cout, int CinPad>
__global__ __launch_bounds__(32*(Cout/16))
void k_conv3x3_wmma(const float* __restrict__ in, const float* __restrict__ wr,
                    const float* __restrict__ bias, float* __restrict__ out) {
  constexpr int TM     = Cout / 16;                    // waves per block
  constexpr int NTHR   = 32 * TM;
  constexpr int CHUNK  = (Cin < 64) ? Cin : 64;        // real channels / chunk
  constexpr int SPAD   = (CHUNK + 3) & ~3;             // staged (padded) chans
  constexpr int CSTC   = (SPAD % 64 == 0) ? SPAD + 4 : SPAD;  // 52 or 68
  constexpr int NCHUNK = (Cin + 63) / 64;

  __shared__ float smem[198 * CSTC];                   // 198 pixel slots

  int lane   = threadIdx.x & 31;
  int m_tile = threadIdx.x >> 5;
  int seg    = blockIdx.x;                             // 64-pixel segment
  int n    = lane & 15;
  int half = lane >> 4;
  int p0    = seg * 64;
  int b     = p0 >> 14;
  int p_im0 = p0 & (HWSZ - 1);
  int y    = p_im0 >> 7;
  int xseg = p_im0 & 127;                              // 0 or 64

  const float* wrow = wr + (size_t)(m_tile*16 + n) * 9 * CinPad;  // A row
  const float* inb  = in + (size_t)b * Cin * HWSZ;
  const float* sbase = smem + n * CSTC;                // lane pixel base

  v8f a0 = {0.f,0.f,0.f,0.f,0.f,0.f,0.f,0.f};
  v8f a1 = a0, a2 = a0, a3 = a0;

  for (int chunk = 0; chunk < NCHUNK; ++chunk) {
    const int ci0 = chunk * 64;

    if (chunk) __syncthreads();
    // ---- stage: pixel-per-thread, channel-inner writes; zero halo + pad ----
    for (int ps = threadIdx.x; ps < 198; ps += NTHR) {
      int r = ps / 66;
      int p = ps - r*66;
      int yy = y - 1 + r;
      int xx = xseg - 1 + p;
      bool ok = (yy >= 0 && yy < HH && xx >= 0 && xx < WW);
      const float* src = inb + (size_t)ci0*HWSZ + yy*WW + xx;
      float* dst = smem + ps * CSTC;
#pragma unroll 8
      for (int ci = 0; ci < CHUNK; ++ci)
        dst[ci] = ok ? src[(size_t)ci * HWSZ] : 0.f;
#pragma unroll
      for (int ci = CHUNK; ci < SPAD; ++ci)            // zero pad channels
        dst[ci] = 0.f;
    }
    __syncthreads();

    // ---- compute: ci-blocks of 4; lane = channels cb*4 + 2*half + {0,1} ----
    for (int cb = 0; cb < SPAD/4; ++cb) {
      int ciA = cb*4 + half*2;
      step9<CinPad, CSTC>(wrow + ci0 + ciA, sbase + ciA, a0, a1, a2, a3);
    }
  }

  // ---- store: D VGPR r -> co = m_tile*16 + 8*half + r, pixel = p_im0+16j+n
  float bs[8];
#pragma unroll
  for (int r = 0; r < 8; ++r) bs[r] = bias[m_tile*16 + half*8 + r];
#pragma unroll
  for (int r = 0; r < 8; ++r) {
    int co = m_tile*16 + half*8 + r;
    float* op = out + ((size_t)(b*Cout + co))*HWSZ + p_im0 + n;
    float v0 = a0[r] + bs[r]; v0 = v0 >= 0.f ? v0 : 0.1f*v0;
    float v1 = a1[r] + bs[r]; v1 = v1 >= 0.f ? v1 : 0.1f*v1;
    float v2 = a2[r] + bs[r]; v2 = v2 >= 0.f ? v2 : 0.1f*v2;
    float v3 = a3[r] + bs[r]; v3 = v3 >= 0.f ? v3 : 0.1f*v3;
    op[0] = v0; op[16] = v1; op[32] = v2; op[48] = v3;
  }
}

// ---------------------------------------------------------------------------
// 5) Final 32->2 conv (no activation) + residual flow_up
// ---------------------------------------------------------------------------
__global__ void k_conv4_add(const float* __restrict__ h3, const float* __restrict__ w4,
                            const float* __restrict__ b4, const float* __restrict__ flow_up,
                            float* __restrict__ out) {
  int idx = blockIdx.x * blockDim.x + threadIdx.x;    // 131072
  if (idx >= BATCH*2*HWSZ) return;
  int x = idx & 127, y = (idx >> 7) & 127, g = (idx >> 14) & 1, b = idx >> 15;
  float acc = b4[g];
  for (int ci = 0; ci < 32; ++ci) {
#pragma unroll
    for (int ky = 0; ky < 3; ++ky) {
      int yy = y + ky - 1;
      if (yy < 0 || yy >= HH) continue;
#pragma unroll
      for (int kx = 0; kx < 3; ++kx) {
        int xx = x + kx - 1;
        if (xx < 0 || xx >= WW) continue;
        acc += w4[((g*32 + ci)*3 + ky)*3 + kx] * h3[(((size_t)b*32 + ci)*HH + yy)*WW + xx];
      }
    }
  }
  out[idx] = flow_up[idx] + acc;
}

// ---------------------------------------------------------------------------
extern "C" void kernel_launch(void* const* d_in, const int* in_sizes, int n_in,
                              void* d_out, int out_size, void* d_ws, size_t ws_size,
                              hipStream_t stream) {
  const float* feature1 = (const float*)d_in[0];
  const float* feature2 = (const float*)d_in[1];
  const float* flow     = (const float*)d_in[2];
  const float* up_w     = (const float*)d_in[3];
  const float* w1 = (const float*)d_in[4];  const float* b1 = (const float*)d_in[5];
  const float* w2 = (const float*)d_in[6];  const float* b2 = (const float*)d_in[7];
  const float* w3 = (const float*)d_in[8];  const float* b3 = (const float*)d_in[9];
  const float* w4 = (const float*)d_in[10]; const float* b4 = (const float*)d_in[11];
  float* outp = (float*)d_out;

  char* wsb = (char*)d_ws;
  const size_t SZ_FLOWUP = (size_t)BATCH*2*HWSZ*4;        // 0.5 MB
  const size_t SZ_FEAT   = (size_t)BATCH*CH*HWSZ*4;       // 33.5 MB
  const size_t SZ_CORR   = (size_t)BATCH*49*HWSZ*4;       // 12.8 MB
  const size_t SZ_WR1    = (size_t)128*9*52*4;
  const size_t SZ_WR2    = (size_t)64*9*128*4;
  const size_t SZ_WR3    = (size_t)32*9*64*4;
  float* flow_up = (float*)(wsb);
  float* bufA    = (float*)(wsb + SZ_FLOWUP);             // f2w, later h2 (64ch)
  float* bufC    = (float*)(wsb + SZ_FLOWUP + SZ_FEAT);   // corr, later h3 (32ch)
  float* bufB    = (float*)(wsb + SZ_FLOWUP + SZ_FEAT + SZ_CORR);      // h1
  float* wr1     = (float*)(wsb + SZ_FLOWUP + 2*SZ_FEAT + SZ_CORR);
  float* wr2     = (float*)((char*)wr1 + SZ_WR1);
  float* wr3     = (float*)((char*)wr2 + SZ_WR2);

  // weight reorders (independent, launch first)
  k_wreorder<49,128,52> <<<(128*9*52 +255)/256, 256, 0, stream>>>(w1, wr1);
  k_wreorder<128,64,128><<<(64*9*128 +255)/256, 256, 0, stream>>>(w2, wr2);
  k_wreorder<64,32,64>  <<<(32*9*64  +255)/256, 256, 0, stream>>>(w3, wr3);

  k_upsample<<<512, 256, 0, stream>>>(flow, up_w, flow_up);
  k_warp<<<dim3(HH, BATCH), WW, 0, stream>>>(feature2, flow_up, bufA);
  k_corr<<<dim3(HH, BATCH), WW, 0, stream>>>(feature1, bufA, bufC);

  // WMMA conv stack: 1024 blocks (one per 64-pixel segment), waves = Cout/16
  const int segs = BATCH*HWSZ/64;   // 1024
  k_conv3x3_wmma<49,128,52> <<<segs, 32*(128/16), 0, stream>>>(bufC, wr1, b1, bufB);
  k_conv3x3_wmma<128,64,128><<<segs, 32*(64/16),  0, stream>>>(bufB, wr2, b2, bufA);
  k_conv3x3_wmma<64,32,64>  <<<segs, 32*(32/16),  0, stream>>>(bufA, wr3, b3, bufC);

  k_conv4_add<<<512, 256, 0, stream>>>(bufC, w4, b4, flow_up, outp);
}